// FlexibleAttention_10557029614163
// MI455X (gfx1250) — compile-verified
//
#include <hip/hip_runtime.h>
#include <hip/hip_bf16.h>
#include <math.h>

typedef __attribute__((ext_vector_type(2))) float v2f;
typedef __attribute__((ext_vector_type(8))) float v8f;

#define EPS_RHO 1e-8f
#define EPS_DEN 1e-9f

__device__ __forceinline__ float rho_relu(float x) {
    float p = sqrtf(fmaxf(x, 0.f) + EPS_RHO);
    float n = sqrtf(fmaxf(-x, 0.f) + EPS_RHO);
    return fmaxf(p - n, 0.f);
}

__device__ __forceinline__ v8f wmma_f32(v2f a, v2f b, v8f c) {
    return __builtin_amdgcn_wmma_f32_16x16x4_f32(false, a, false, b,
                                                 (short)0, c, false, false);
}

// ---------------------------------------------------------------------------
// Out[M x Nc] = A[M x K] * W[Nc x K]^T (+ bias[n]).
// Register-blocked: each wave owns a 2x4 grid of 16x16 tiles (32 rows x 64
// cols).  Six streaming pointers (2 A + 4 B) are advanced per K-step and all
// six fragments are loaded before any WMMA, so the loads clause together and
// drain across the 8 WMMAs instead of serializing on s_wait_loadcnt 0.
// A-fragment layout (f32 16x16x4): lane = M row (lane&15), VGPR0/1 = K{0,1}
// for lanes 0-15, K{2,3} for lanes 16-31 (ISA 7.12.2); B mirrored, lane = N.
// ---------------------------------------------------------------------------
__global__ void fa_gemm_abt(const float* __restrict__ A,
                            const float* __restrict__ W,
                            const float* __restrict__ bias,
                            float* __restrict__ Out,
                            int M, int K, int Nc)
{
    const int lane  = threadIdx.x & 31;
    const int wave  = threadIdx.x >> 5;
    const int nblkN = Nc >> 6;                       // 64-col blocks
    const int job   = blockIdx.x * (blockDim.x >> 5) + wave;
    if (job >= (M >> 5) * nblkN) return;             // wave-uniform exit
    const int mb = job / nblkN, nb = job % nblkN;
    const int l15  = lane & 15;
    const int koff = (lane < 16) ? 0 : 2;
    const size_t bstep = (size_t)16 * K;

    const float* pA0 = A + (size_t)((mb << 5) + l15) * K + koff;
    const float* pA1 = pA0 + bstep;
    const float* pB0 = W + (size_t)((nb << 6) + l15) * K + koff;
    const float* pB1 = pB0 + bstep;
    const float* pB2 = pB1 + bstep;
    const float* pB3 = pB2 + bstep;

    v8f acc[2][4] = {};
    #pragma unroll 2
    for (int kk = 0; kk < K; kk += 4) {
        // load all fragments first -> independent, clause-friendly
        const v2f a0 = *(const v2f*)pA0;
        const v2f a1 = *(const v2f*)pA1;
        const v2f b0 = *(const v2f*)pB0;
        const v2f b1 = *(const v2f*)pB1;
        const v2f b2 = *(const v2f*)pB2;
        const v2f b3 = *(const v2f*)pB3;
        pA0 += 4; pA1 += 4; pB0 += 4; pB1 += 4; pB2 += 4; pB3 += 4;

        acc[0][0] = wmma_f32(a0, b0, acc[0][0]);
        acc[1][0] = wmma_f32(a1, b0, acc[1][0]);
        acc[0][1] = wmma_f32(a0, b1, acc[0][1]);
        acc[1][1] = wmma_f32(a1, b1, acc[1][1]);
        acc[0][2] = wmma_f32(a0, b2, acc[0][2]);
        acc[1][2] = wmma_f32(a1, b2, acc[1][2]);
        acc[0][3] = wmma_f32(a0, b3, acc[0][3]);
        acc[1][3] = wmma_f32(a1, b3, acc[1][3]);
    }

    const int mbase = (mb << 5) + ((lane < 16) ? 0 : 8);
    #pragma unroll
    for (int ni = 0; ni < 4; ++ni) {
        const int   n  = (nb << 6) + (ni << 4) + l15;
        const float bn = bias ? bias[n] : 0.f;
        #pragma unroll
        for (int mi = 0; mi < 2; ++mi)
            #pragma unroll
            for (int r = 0; r < 8; ++r)
                Out[(size_t)(mbase + (mi << 4) + r) * Nc + n] = acc[mi][ni][r] + bn;
    }
}

// ---------------------------------------------------------------------------
// Per (b,i) row: scores_ij = sum_a relu(rho(q_ia + k_ja)) * Wphi_a,
// then ex = exp(s - rowmax) stored to ws, denom = sum(ex).
// 256 threads / block, each thread owns j = tid and tid+256.
// ---------------------------------------------------------------------------
__global__ void fa_scores(const float* __restrict__ q,
                          const float* __restrict__ k,
                          const float* __restrict__ Wphi,
                          float* __restrict__ ex,
                          float* __restrict__ denom)
{
    __shared__ float qi[64];
    __shared__ float wph[64];
    __shared__ float red[8];
    __shared__ float msh;
    const int row = blockIdx.x;           // 0..1023  (b*512 + i)
    const int b   = row >> 9;
    const int tid = threadIdx.x;
    if (tid < 64) {
        qi[tid]  = q[(size_t)row * 64 + tid];
        wph[tid] = Wphi[tid];
    }
    __syncthreads();

    float s[2];
    #pragma unroll
    for (int p = 0; p < 2; ++p) {
        const int j = tid + (p << 8);
        const float* kj = k + ((size_t)(b * 512 + j)) * 64;
        float acc = 0.f;
        #pragma unroll 8
        for (int a = 0; a < 64; ++a)
            acc += rho_relu(qi[a] + kj[a]) * wph[a];
        s[p] = acc;
    }

    // block-max
    float mymax = fmaxf(s[0], s[1]);
    for (int off = 16; off; off >>= 1)
        mymax = fmaxf(mymax, __shfl_xor(mymax, off));
    const int lane = tid & 31, wv = tid >> 5;
    if (lane == 0) red[wv] = mymax;
    __syncthreads();
    if (tid == 0) {
        float m = red[0];
        for (int w = 1; w < 8; ++w) m = fmaxf(m, red[w]);
        msh = m;
    }
    __syncthreads();
    const float m = msh;

    const float e0 = expf(s[0] - m);
    const float e1 = expf(s[1] - m);
    ex[(size_t)row * 512 + tid]       = e0;
    ex[(size_t)row * 512 + tid + 256] = e1;

    float mysum = e0 + e1;
    for (int off = 16; off; off >>= 1)
        mysum += __shfl_xor(mysum, off);
    if (lane == 0) red[wv] = mysum;
    __syncthreads();
    if (tid == 0) {
        float sm = 0.f;
        for (int w = 0; w < 8; ++w) sm += red[w];
        denom[row] = sm;
    }
}

// ---------------------------------------------------------------------------
// part1[b,i,c] = sum_j ex[b,i,j] * Vy[b,j,c]   (per-batch 512x512 @ 512x1024)
// Register-blocked 2x4 tiles per wave; all 10 operand loads per K-step are
// issued before the 8 WMMAs (streaming pointers, no per-tile address math).
// ---------------------------------------------------------------------------
__global__ void fa_part1(const float* __restrict__ ex,
                         const float* __restrict__ Vy,
                         float* __restrict__ part1)
{
    const int lane = threadIdx.x & 31;
    const int wave = threadIdx.x >> 5;
    const int job  = blockIdx.x * (blockDim.x >> 5) + wave; // 512 jobs total
    const int b    = job >> 8;                              // 256 jobs / batch
    const int jj   = job & 255;
    const int mb   = jj >> 4;                               // 16 row-blocks (32 rows)
    const int nb   = jj & 15;                               // 16 col-blocks (64 cols)
    const int l15  = lane & 15;
    const int koff = (lane < 16) ? 0 : 2;
    const int col0 = (nb << 6) + l15;

    const float* pA0 = ex + ((size_t)(b * 512 + (mb << 5) + l15)) * 512 + koff;
    const float* pA1 = pA0 + (size_t)16 * 512;
    // B rows j+koff and j+koff+1 of Vy, four 16-col tiles
    const float* pV  = Vy + (size_t)b * 512 * 1024 + (size_t)koff * 1024 + col0;

    v8f acc[2][4] = {};
    #pragma unroll 2
    for (int j = 0; j < 512; j += 4) {
        const v2f a0 = *(const v2f*)pA0;
        const v2f a1 = *(const v2f*)pA1;
        v2f b0, b1, b2, b3;
        b0.x = pV[0];          b0.y = pV[1024];
        b1.x = pV[16];         b1.y = pV[1024 + 16];
        b2.x = pV[32];         b2.y = pV[1024 + 32];
        b3.x = pV[48];         b3.y = pV[1024 + 48];
        pA0 += 4; pA1 += 4; pV += 4 * 1024;

        acc[0][0] = wmma_f32(a0, b0, acc[0][0]);
        acc[1][0] = wmma_f32(a1, b0, acc[1][0]);
        acc[0][1] = wmma_f32(a0, b1, acc[0][1]);
        acc[1][1] = wmma_f32(a1, b1, acc[1][1]);
        acc[0][2] = wmma_f32(a0, b2, acc[0][2]);
        acc[1][2] = wmma_f32(a1, b2, acc[1][2]);
        acc[0][3] = wmma_f32(a0, b3, acc[0][3]);
        acc[1][3] = wmma_f32(a1, b3, acc[1][3]);
    }

    const int mbase = b * 512 + (mb << 5) + ((lane < 16) ? 0 : 8);
    #pragma unroll
    for (int ni = 0; ni < 4; ++ni) {
        const int n = col0 + (ni << 4);
        #pragma unroll
        for (int mi = 0; mi < 2; ++mi)
            #pragma unroll
            for (int r = 0; r < 8; ++r)
                part1[(size_t)(mbase + (mi << 4) + r) * 1024 + n] = acc[mi][ni][r];
    }
}

// ---------------------------------------------------------------------------
// Ew[row,a] = sum_j ex[row,j] * relu(rho(q[row,a] + k[b*512+j, a]))
// 64 threads / block (lane = a), ex row staged through LDS, k reads coalesced.
// ---------------------------------------------------------------------------
__global__ void fa_ew(const float* __restrict__ q,
                      const float* __restrict__ k,
                      const float* __restrict__ ex,
                      float* __restrict__ Ew)
{
    __shared__ float exs[512];
    const int row = blockIdx.x;
    const int b   = row >> 9;
    const int a   = threadIdx.x;          // 0..63
    for (int j = a; j < 512; j += 64)
        exs[j] = ex[(size_t)row * 512 + j];
    __syncthreads();
    const float qa = q[(size_t)row * 64 + a];
    float acc = 0.f;
    #pragma unroll 4
    for (int j = 0; j < 512; ++j) {
        const float e = rho_relu(qa + k[((size_t)(b * 512 + j)) * 64 + a]);
        acc += exs[j] * e;
    }
    Ew[(size_t)row * 64 + a] = acc;
}

// ---------------------------------------------------------------------------
// out = (part1 + Ew @ We_w^T + We_b*denom) / (denom + EPS_DEN)
// K = 64 WMMA GEMM with fused epilogue.
// ---------------------------------------------------------------------------
__global__ void fa_final(const float* __restrict__ Ew,
                         const float* __restrict__ Wew,   // 1024 x 64
                         const float* __restrict__ Web,   // 1024
                         const float* __restrict__ part1,
                         const float* __restrict__ denom,
                         float* __restrict__ out)
{
    const int lane = threadIdx.x & 31;
    const int wave = threadIdx.x >> 5;
    const int t  = blockIdx.x * (blockDim.x >> 5) + wave; // 64*64 = 4096 tiles
    const int mt = t >> 6, nt = t & 63;
    const int l15  = lane & 15;
    const int koff = (lane < 16) ? 0 : 2;
    const float* Arow = Ew  + (size_t)((mt << 4) + l15) * 64 + koff;
    const float* Brow = Wew + (size_t)((nt << 4) + l15) * 64 + koff;
    v8f acc = {};
    #pragma unroll
    for (int kk = 0; kk < 64; kk += 4) {
        v2f av = *(const v2f*)(Arow + kk);
        v2f bv = *(const v2f*)(Brow + kk);
        acc = wmma_f32(av, bv, acc);
    }
    const int   n     = (nt << 4) + l15;
    const int   mbase = (mt << 4) + ((lane < 16) ? 0 : 8);
    const float bn    = Web[n];
    #pragma unroll
    for (int r = 0; r < 8; ++r) {
        const int   m = mbase + r;
        const float d = denom[m];
        out[(size_t)m * 1024 + n] =
            (part1[(size_t)m * 1024 + n] + acc[r] + bn * d) / (d + EPS_DEN);
    }
}

// ---------------------------------------------------------------------------
extern "C" void kernel_launch(void* const* d_in, const int* in_sizes, int n_in,
                              void* d_out, int out_size, void* d_ws, size_t ws_size,
                              hipStream_t stream)
{
    (void)in_sizes; (void)n_in; (void)out_size; (void)ws_size;
    const float* X    = (const float*)d_in[0];   // (2,512,1024)
    const float* Wq   = (const float*)d_in[1];   // (64,1024)
    const float* Wk   = (const float*)d_in[2];   // (64,1024)
    const float* Wphi = (const float*)d_in[3];   // (64,)
    const float* Wyw  = (const float*)d_in[4];   // (1024,1024)
    const float* Wyb  = (const float*)d_in[5];   // (1024,)
    const float* Wew  = (const float*)d_in[6];   // (1024,64)
    const float* Web  = (const float*)d_in[7];   // (1024,)
    float* out = (float*)d_out;                  // (2,512,1024)

    // workspace carve-up (~10.8 MB of fp32, all L2-resident on MI455X)
    float* ws  = (float*)d_ws;
    float* q   = ws;  ws += 1024 * 64;
    float* kk  = ws;  ws += 1024 * 64;
    float* Vy  = ws;  ws += 1024 * 1024;
    float* ex  = ws;  ws += 1024 * 512;
    float* den = ws;  ws += 1024;
    float* p1  = ws;  ws += 1024 * 1024;
    float* Ew  = ws;  ws += 1024 * 64;

    // projections (fp32 WMMA, 2x4 tile blocks per wave)
    fa_gemm_abt<<<4,   256, 0, stream>>>(X, Wq,  nullptr, q,  1024, 1024, 64);
    fa_gemm_abt<<<4,   256, 0, stream>>>(X, Wk,  nullptr, kk, 1024, 1024, 64);
    fa_gemm_abt<<<64,  256, 0, stream>>>(X, Wyw, Wyb,     Vy, 1024, 1024, 1024);

    // scores -> ex (2 MB) + denom, fully fused (e never materialized)
    fa_scores<<<1024, 256, 0, stream>>>(q, kk, Wphi, ex, den);

    // part1 = ex @ Vy (fp32 WMMA, per-batch, 2x4 tile blocks): 512 jobs -> 64 blocks
    fa_part1<<<64, 256, 0, stream>>>(ex, Vy, p1);

    // Ew = sum_j ex * e  (streaming recompute of e)
    fa_ew<<<1024, 64, 0, stream>>>(q, kk, ex, Ew);

    // part2 GEMM + epilogue -> out
    fa_final<<<512, 256, 0, stream>>>(Ew, Wew, Web, p1, den, out);
}